// GNN_passing_48266842472727
// MI455X (gfx1250) — compile-verified
//
#include <hip/hip_runtime.h>
#include <hip/hip_bf16.h>

// CDNA5 / gfx1250. wave32. Dense GEMMs via V_WMMA_F32_16X16X4_F32 (full f32,
// workload is memory-bound so no reason to drop precision).

typedef float v2f __attribute__((ext_vector_type(2)));
typedef float v8f __attribute__((ext_vector_type(8)));

#define D 64

// ---------------------------------------------------------------- utilities

__global__ void zero_kernel(float* __restrict__ p, long n) {
    long i = (long)blockIdx.x * blockDim.x + threadIdx.x;
    if (i < n) p[i] = 0.0f;
}

// out[i, d] = root[i, d] + bias[d]
__global__ void init_root_bias_kernel(const float* __restrict__ root,
                                      const float* __restrict__ bias,
                                      float* __restrict__ out, long nd) {
    long i = (long)blockIdx.x * blockDim.x + threadIdx.x;
    if (i < nd) out[i] = root[i] + bias[i & (D - 1)];
}

// cat[i, 0:64] = a1+a2 ; cat[i, 64:128] = b1+b2
__global__ void cat_sum_kernel(const float* __restrict__ a1, const float* __restrict__ a2,
                               const float* __restrict__ b1, const float* __restrict__ b2,
                               float* __restrict__ cat, long n /* = N*128 */) {
    long i = (long)blockIdx.x * blockDim.x + threadIdx.x;
    if (i >= n) return;
    long node = i >> 7;
    int  d    = (int)(i & 127);
    float v;
    if (d < D) {
        long j = node * D + d;
        v = a1[j] + a2[j];
    } else {
        long j = node * D + (d - D);
        v = b1[j] + b2[j];
    }
    cat[i] = v;
}

// ---------------------------------------------------------------- edge passes

// c[dst, r] += 1
__global__ void edge_count_kernel(const int* __restrict__ dst,
                                  const int* __restrict__ etype,
                                  float* __restrict__ cnt, int E, int R) {
    long e = (long)blockIdx.x * blockDim.x + threadIdx.x;
    if (e >= E) return;
    atomicAdd(&cnt[(long)dst[e] * R + etype[e]], 1.0f);
}

// out[dst] += table_row(src, r) / max(cnt[dst,r], 1)
// layer-1: table = W1 [R,N,D], relStride = N*D, rsel = -1 (all relations)
// layer-2: table = Yr [N,D],   relStride = 0,   rsel = r  (filter)
// 16 threads per edge, float4 per thread -> contiguous 256B per 16-lane cluster.
__global__ void scatter_mean_kernel(const float* __restrict__ table,
                                    const int* __restrict__ src,
                                    const int* __restrict__ dst,
                                    const int* __restrict__ etype,
                                    const float* __restrict__ cnt,
                                    float* __restrict__ out,
                                    int E, int R, int rsel, long relStride) {
    long gid = (long)blockIdx.x * blockDim.x + threadIdx.x;
    long e = gid >> 4;
    if (e >= E) return;
    int r = etype[e];
    if (rsel >= 0 && r != rsel) return;
    int s = src[e];
    int d = dst[e];
    float c = cnt[(long)d * R + r];
    float inv = 1.0f / fmaxf(c, 1.0f);
    int d0 = (int)(gid & 15) * 4;
    const float* row = table + (rsel >= 0 ? 0 : (long)r * relStride) + (long)s * D + d0;
    __builtin_prefetch(row, 0, 1);                      // global_prefetch_b8
    float4 v = *(const float4*)row;
    float* o = out + (long)d * D + d0;
    atomicAdd(o + 0, v.x * inv);
    atomicAdd(o + 1, v.y * inv);
    atomicAdd(o + 2, v.z * inv);
    atomicAdd(o + 3, v.w * inv);
}

// ---------------------------------------------------------------- ATTITU head

// One wave (32 lanes) per node; lane l covers features l and l+32.
__global__ void attitu_kernel(const float* __restrict__ e1, const float* __restrict__ e2,
                              const float* __restrict__ w11, const float* __restrict__ b11,
                              const float* __restrict__ w12, const float* __restrict__ b12,
                              const float* __restrict__ w21, const float* __restrict__ b21,
                              const float* __restrict__ w22, const float* __restrict__ b22,
                              float* __restrict__ f1, float* __restrict__ f2, int N) {
    int wavesPerBlock = blockDim.x >> 5;
    int node = blockIdx.x * wavesPerBlock + (threadIdx.x >> 5);
    if (node >= N) return;
    int l = threadIdx.x & 31;
    const float* p1 = e1 + (long)node * D;
    const float* p2 = e2 + (long)node * D;
    float x1a = p1[l], x1b = p1[l + 32];
    float x2a = p2[l], x2b = p2[l + 32];
    float d11 = x1a * w11[l] + x1b * w11[l + 32];
    float d12 = x2a * w12[l] + x2b * w12[l + 32];
    float d21 = x1a * w21[l] + x1b * w21[l + 32];
    float d22 = x2a * w22[l] + x2b * w22[l + 32];
    #pragma unroll
    for (int off = 16; off; off >>= 1) {               // wave32 butterfly
        d11 += __shfl_xor(d11, off, 32);
        d12 += __shfl_xor(d12, off, 32);
        d21 += __shfl_xor(d21, off, 32);
        d22 += __shfl_xor(d22, off, 32);
    }
    d11 += b11[0]; d12 += b12[0]; d21 += b21[0]; d22 += b22[0];
    float m1 = fmaxf(d11, d12);
    float a1 = __expf(d11 - m1), a2 = __expf(d12 - m1);
    float inv1 = 1.0f / (a1 + a2);
    float s10 = a1 * inv1, s11 = a2 * inv1;
    float m2 = fmaxf(d21, d22);
    float g1 = __expf(d21 - m2), g2 = __expf(d22 - m2);
    float inv2 = 1.0f / (g1 + g2);
    float s20 = g1 * inv2, s21 = g2 * inv2;
    float* q1 = f1 + (long)node * D;
    float* q2 = f2 + (long)node * D;
    q1[l]      = s10 * x1a + s11 * x2a;
    q1[l + 32] = s10 * x1b + s11 * x2b;
    q2[l]      = s20 * x1a + s21 * x2a;
    q2[l + 32] = s20 * x1b + s21 * x2b;
}

// ---------------------------------------------------------------- WMMA GEMM

// C[N,64] = A[N,K] @ B[K,64] (+ bias) (opt. LeakyReLU(0.01)); full f32 via
// V_WMMA_F32_16X16X4_F32. One wave per 16x16 output tile; N % 16 == 0.
// Fragment layout (ISA 7.12.2):
//   A 16x4: lanes 0-15 -> K = {k,k+1}, lanes 16-31 -> K = {k+2,k+3}, M = lane&15
//   B 4x16 / C 16x16: row-striped across lanes; lanes 0-15 N=lane, rows M..M+7
__global__ void wmma_gemm_n64_kernel(const float* __restrict__ A,
                                     const float* __restrict__ B,
                                     const float* __restrict__ bias,
                                     float* __restrict__ C,
                                     int N, int K, int flags /* bit0: leaky */) {
    int wave = blockIdx.x * (blockDim.x >> 5) + (threadIdx.x >> 5);
    int rowTiles = N >> 4;
    int colTile = wave & 3;            // 64 cols = 4 tiles
    int rowTile = wave >> 2;
    if (rowTile >= rowTiles) return;
    int lane = threadIdx.x & 31;
    int m    = lane & 15;
    int khalf = (lane >> 4) << 1;      // 0 or 2
    int row0 = rowTile << 4;
    int col0 = colTile << 4;

    v8f acc = {};
    const float* arow = A + (long)(row0 + m) * K + khalf;
    for (int k = 0; k < K; k += 4) {
        v2f a, b;
        a.x = arow[k];
        a.y = arow[k + 1];
        const float* bp = B + (long)(k + khalf) * D + col0 + m;
        b.x = bp[0];
        b.y = bp[D];
        acc = __builtin_amdgcn_wmma_f32_16x16x4_f32(
            false, a, false, b, (short)0, acc, false, false);
    }

    int n = col0 + m;
    int mrow = row0 + ((lane >> 4) << 3);   // lanes 16-31 hold rows +8..+15
    float bv = bias ? bias[n] : 0.0f;
    #pragma unroll
    for (int i = 0; i < 8; ++i) {
        float v = acc[i] + bv;
        if (flags & 1) v = v > 0.0f ? v : 0.01f * v;
        C[(long)(mrow + i) * D + n] = v;
    }
}

// ---------------------------------------------------------------- host side

static inline int blocks_for(long n, int bs) { return (int)((n + bs - 1) / bs); }

static void launch_gemm(const float* A, const float* B, const float* bias, float* C,
                        int N, int K, int flags, hipStream_t s) {
    long waves = (long)(N / 16) * 4;
    int blocks = blocks_for(waves * 32, 256);
    wmma_gemm_n64_kernel<<<blocks, 256, 0, s>>>(A, B, bias, C, N, K, flags);
}

extern "C" void kernel_launch(void* const* d_in, const int* in_sizes, int n_in,
                              void* d_out, int out_size, void* d_ws, size_t ws_size,
                              hipStream_t stream) {
    // -------- inputs (setup_inputs order)
    const int*   ei_ex   = (const int*)d_in[0];
    const int*   et_ex   = (const int*)d_in[1];
    const int*   ei_im   = (const int*)d_in[2];
    const int*   et_im   = (const int*)d_in[3];
    const float* W1_ex   = (const float*)d_in[4];
    const float* root1ex = (const float*)d_in[5];
    const float* b1_ex   = (const float*)d_in[6];
    const float* W1_im   = (const float*)d_in[7];
    const float* root1im = (const float*)d_in[8];
    const float* b1_im   = (const float*)d_in[9];
    const float* W2_ex   = (const float*)d_in[10];
    const float* root2ex = (const float*)d_in[11];
    const float* b2_ex   = (const float*)d_in[12];
    const float* W2_im   = (const float*)d_in[13];
    const float* root2im = (const float*)d_in[14];
    const float* b2_im   = (const float*)d_in[15];
    const float* aw11 = (const float*)d_in[16]; const float* ab11 = (const float*)d_in[17];
    const float* aw12 = (const float*)d_in[18]; const float* ab12 = (const float*)d_in[19];
    const float* aw21 = (const float*)d_in[20]; const float* ab21 = (const float*)d_in[21];
    const float* aw22 = (const float*)d_in[22]; const float* ab22 = (const float*)d_in[23];
    const float* aggW1 = (const float*)d_in[24]; const float* aggb1 = (const float*)d_in[25];
    const float* aggW2 = (const float*)d_in[26]; const float* aggb2 = (const float*)d_in[27];

    const int N    = in_sizes[5] / D;          // root1_ex is [N, D]
    const int E_EX = in_sizes[1];
    const int E_IM = in_sizes[3];
    const int R_EX = in_sizes[4] / (N * D);    // W1_ex is [R_EX, N, D]
    const int R_IM = in_sizes[7] / (N * D);
    const long ND  = (long)N * D;

    const int* src_ex = ei_ex;            const int* dst_ex = ei_ex + E_EX;
    const int* src_im = ei_im;            const int* dst_im = ei_im + E_IM;

    // -------- workspace layout (~261 MB total)
    float* ws     = (float*)d_ws;
    float* cnt_ex = ws;                      // N * R_EX
    float* cnt_im = cnt_ex + (long)N * R_EX; // N * R_IM
    float* ex_x1  = cnt_im + (long)N * R_IM;
    float* im_x1  = ex_x1 + ND;
    float* t_ex   = im_x1 + ND;
    float* t_im   = t_ex + ND;
    float* ex_x2  = t_im + ND;
    float* im_x2  = ex_x2 + ND;
    float* catb   = im_x2 + ND;              // N * 2D
    float* Hbuf   = catb + 2 * ND;           // N * D
    float* Yr     = Hbuf + ND;               // N * D (per-relation, reused)

    // -------- layer 1: counts, root+bias init, scatter-mean of W1[r, src]
    long ncnt = (long)N * (R_EX + R_IM);
    zero_kernel<<<blocks_for(ncnt, 256), 256, 0, stream>>>(cnt_ex, ncnt);
    edge_count_kernel<<<blocks_for(E_EX, 256), 256, 0, stream>>>(dst_ex, et_ex, cnt_ex, E_EX, R_EX);
    edge_count_kernel<<<blocks_for(E_IM, 256), 256, 0, stream>>>(dst_im, et_im, cnt_im, E_IM, R_IM);

    init_root_bias_kernel<<<blocks_for(ND, 256), 256, 0, stream>>>(root1ex, b1_ex, ex_x1, ND);
    init_root_bias_kernel<<<blocks_for(ND, 256), 256, 0, stream>>>(root1im, b1_im, im_x1, ND);

    scatter_mean_kernel<<<blocks_for((long)E_EX * 16, 256), 256, 0, stream>>>(
        W1_ex, src_ex, dst_ex, et_ex, cnt_ex, ex_x1, E_EX, R_EX, -1, ND);
    scatter_mean_kernel<<<blocks_for((long)E_IM * 16, 256), 256, 0, stream>>>(
        W1_im, src_im, dst_im, et_im, cnt_im, im_x1, E_IM, R_IM, -1, ND);

    // -------- ATTITU fusion -> t_ex, t_im   (8 waves / block, wave per node)
    attitu_kernel<<<blocks_for(N, 8), 256, 0, stream>>>(
        ex_x1, im_x1, aw11, ab11, aw12, ab12, aw21, ab21, aw22, ab22, t_ex, t_im, N);

    // -------- layer 2 explicit: x2 = t@root + b, then += mean_r(t[src]) @ W2[r]
    // Using linearity: precompute Y_r = t @ W2[r] (WMMA), then reuse scatter-mean.
    launch_gemm(t_ex, root2ex, b2_ex, ex_x2, N, D, 0, stream);
    for (int r = 0; r < R_EX; ++r) {
        launch_gemm(t_ex, W2_ex + (long)r * D * D, nullptr, Yr, N, D, 0, stream);
        scatter_mean_kernel<<<blocks_for((long)E_EX * 16, 256), 256, 0, stream>>>(
            Yr, src_ex, dst_ex, et_ex, cnt_ex, ex_x2, E_EX, R_EX, r, 0);
    }
    // -------- layer 2 implicit
    launch_gemm(t_im, root2im, b2_im, im_x2, N, D, 0, stream);
    for (int r = 0; r < R_IM; ++r) {
        launch_gemm(t_im, W2_im + (long)r * D * D, nullptr, Yr, N, D, 0, stream);
        scatter_mean_kernel<<<blocks_for((long)E_IM * 16, 256), 256, 0, stream>>>(
            Yr, src_im, dst_im, et_im, cnt_im, im_x2, E_IM, R_IM, r, 0);
    }

    // -------- head: cat = [ex_x1+ex_x2, im_x1+im_x2]; MLP with WMMA
    cat_sum_kernel<<<blocks_for(2 * ND, 256), 256, 0, stream>>>(
        ex_x1, ex_x2, im_x1, im_x2, catb, 2 * ND);
    launch_gemm(catb, aggW1, aggb1, Hbuf, N, 2 * D, /*leaky*/ 1, stream);
    launch_gemm(Hbuf, aggW2, aggb2, (float*)d_out, N, D, 0, stream);
}